// DeformableTransformerEncoderLayer_31009663877488
// MI455X (gfx1250) — compile-verified
//
#include <hip/hip_runtime.h>
#include <hip/hip_bf16.h>
#include <math.h>

typedef __attribute__((ext_vector_type(2))) float v2f;
typedef __attribute__((ext_vector_type(8))) float v8f;

#define D_      256
#define NH_     8
#define HD_     32
#define NL_     4
#define NP_     4
#define DFF_    1024
#define LIN_    21760
#define WAVES_  4            // waves per GEMM block

// ---------------------------------------------------------------------------
// Pack W[K, NOUT] into K-pair-major: Wp[(k/2)*NOUT + n] = {W[k][n], W[k+1][n]}
// so a WMMA B fragment is one aligned b64 load per lane.
// ---------------------------------------------------------------------------
__global__ __launch_bounds__(256) void pack_w_kernel(
    const float* __restrict__ W, float* __restrict__ Wp, int total, int NOUT)
{
    const int idx = blockIdx.x * blockDim.x + threadIdx.x;
    if (idx >= total) return;
    const int k = idx / NOUT;
    const int n = idx - k * NOUT;
    Wp[((size_t)(k >> 1) * NOUT + n) * 2 + (k & 1)] = W[idx];
}

// ---------------------------------------------------------------------------
// GEMM: Y[Ntok, NOUT] = (X (+X2)) @ W[K, NOUT] + bias  (optional ReLU)
// W pre-packed (pair-major). One wave owns a (RT*16) x (CT*16) slab:
// RT*CT f32-WMMA accumulators; per k-step: RT A-b64 + CT B-b64 -> RT*CT WMMAs.
// Compile-time K/NOUT -> immediate addressing.
// ---------------------------------------------------------------------------
template <bool ADD2, bool RELU, int RT, int CT, int K, int NOUT>
__global__ __launch_bounds__(32 * WAVES_) void gemm_wmma(
    const float* __restrict__ X, const float* __restrict__ X2,
    const float* __restrict__ Wp, const float* __restrict__ bias,
    float* __restrict__ Y)
{
    const int lane  = threadIdx.x & 31;
    const int wave  = threadIdx.x >> 5;
    const int rbase = blockIdx.x * (RT * 16);                    // row slab
    const int cbase = (blockIdx.y * WAVES_ + wave) * (CT * 16);  // column slab
    const int m     = lane & 15;
    const int koff  = lane >> 4;                                 // 0/1: K sub-pair
    const int n     = m;

    const float* __restrict__ xrow  = X + (size_t)(rbase + m) * K;
    const float* __restrict__ x2row = ADD2 ? (X2 + (size_t)(rbase + m) * K) : nullptr;
    const v2f*   __restrict__ wp2   = (const v2f*)Wp + cbase + n;  // lane column base

    v8f acc[RT][CT];
#pragma unroll
    for (int rt = 0; rt < RT; ++rt)
#pragma unroll
        for (int ct = 0; ct < CT; ++ct) acc[rt][ct] = (v8f){};

    for (int kk = 0; kk < K; kk += 32) {
        if (kk + 32 < K) {
#pragma unroll
            for (int rt = 0; rt < RT; ++rt) {
                __builtin_prefetch(xrow + (size_t)rt * 16 * K + kk + 32, 0, 3);
                if (ADD2) __builtin_prefetch(x2row + (size_t)rt * 16 * K + kk + 32, 0, 3);
            }
        }
#pragma unroll
        for (int k4 = 0; k4 < 32; k4 += 4) {
            const int k  = kk + k4;
            const int ka = k + 2 * koff;          // lane's K pair start
            // A fragments: lane holds A[row][ka], A[row][ka+1] (contiguous b64)
            v2f a[RT];
#pragma unroll
            for (int rt = 0; rt < RT; ++rt) {
                const size_t ao = (size_t)rt * 16 * K + ka;
                a[rt].x = xrow[ao];
                a[rt].y = xrow[ao + 1];
                if (ADD2) { a[rt].x += x2row[ao]; a[rt].y += x2row[ao + 1]; }
            }
            // B fragments: one b64 each from packed W
            v2f b[CT];
#pragma unroll
            for (int ct = 0; ct < CT; ++ct)
                b[ct] = wp2[(size_t)(k / 2 + koff) * NOUT + ct * 16];
            // RT*CT WMMAs share the fragments
#pragma unroll
            for (int rt = 0; rt < RT; ++rt)
#pragma unroll
                for (int ct = 0; ct < CT; ++ct)
                    acc[rt][ct] = __builtin_amdgcn_wmma_f32_16x16x4_f32(
                        false, a[rt], false, b[ct], (short)0, acc[rt][ct], false, false);
        }
    }

#pragma unroll
    for (int ct = 0; ct < CT; ++ct) {
        const int col = cbase + ct * 16 + n;
        const float bv = bias[col];
#pragma unroll
        for (int rt = 0; rt < RT; ++rt) {
#pragma unroll
            for (int r = 0; r < 8; ++r) {
                const int row = rbase + rt * 16 + r + 8 * koff;  // C/D reg r -> row r+8*half
                float v = acc[rt][ct][r] + bv;
                if (RELU) v = fmaxf(v, 0.f);
                Y[(size_t)row * NOUT + col] = v;
            }
        }
    }
}

// ---------------------------------------------------------------------------
// Softmax over the last dim (16) of attn logits, in place. One thread per row.
// ---------------------------------------------------------------------------
__global__ __launch_bounds__(256) void softmax16_kernel(float* __restrict__ a, int nrows)
{
    const int i = blockIdx.x * blockDim.x + threadIdx.x;
    if (i >= nrows) return;
    float* __restrict__ r = a + (size_t)i * 16;
    float m = r[0];
#pragma unroll
    for (int j = 1; j < 16; ++j) m = fmaxf(m, r[j]);
    float e[16];
    float s = 0.f;
#pragma unroll
    for (int j = 0; j < 16; ++j) { e[j] = __expf(r[j] - m); s += e[j]; }
    const float inv = 1.f / s;
#pragma unroll
    for (int j = 0; j < 16; ++j) r[j] = e[j] * inv;
}

// ---------------------------------------------------------------------------
// Multi-scale deformable attention sampling.
// One wave per (token, head); lane = channel (HD=32) -> coalesced 128B gathers
// from the L2-resident value tensor.
// ---------------------------------------------------------------------------
__global__ __launch_bounds__(256) void msdeform_kernel(
    const float* __restrict__ value, const float* __restrict__ attn,
    const float* __restrict__ offs,  const float* __restrict__ refp,
    float* __restrict__ out, int Ntok, int Lq)
{
    const int lane = threadIdx.x & 31;
    const int wid  = blockIdx.x * (blockDim.x >> 5) + (threadIdx.x >> 5);
    const int head = wid & (NH_ - 1);
    const int tok  = wid >> 3;
    if (tok >= Ntok) return;
    const int b = tok / Lq;

    const int Hs[NL_] = {128, 64, 32, 16};
    const int Ws[NL_] = {128, 64, 32, 16};
    const int st[NL_] = {0, 16384, 20480, 21504};

    const float* __restrict__ ref   = refp + (size_t)tok * (NL_ * 2);
    const float* __restrict__ orow  = offs + (size_t)tok * D_ + head * (NL_ * NP_ * 2);
    const float* __restrict__ arow  = attn + (size_t)tok * (NH_ * NL_ * NP_) + head * (NL_ * NP_);
    const float* __restrict__ vbase = value + (size_t)b * Lq * D_ + head * HD_ + lane;

    float acc = 0.f;
#pragma unroll
    for (int l = 0; l < NL_; ++l) {
        const int Hl = Hs[l], Wl = Ws[l];
        const float rx = ref[l * 2 + 0] * (float)Wl - 0.5f;
        const float ry = ref[l * 2 + 1] * (float)Hl - 0.5f;
        const float* __restrict__ vlev = vbase + (size_t)st[l] * D_;
#pragma unroll
        for (int p = 0; p < NP_; ++p) {
            const float x = rx + orow[(l * NP_ + p) * 2 + 0];
            const float y = ry + orow[(l * NP_ + p) * 2 + 1];
            const float x0f = floorf(x), y0f = floorf(y);
            const float lx = x - x0f, ly = y - y0f;
            const int x0 = (int)x0f, y0 = (int)y0f;
            const int x1 = x0 + 1,  y1 = y0 + 1;
            const float a = arow[l * NP_ + p];
            const float w00 = a * (1.f - lx) * (1.f - ly);
            const float w10 = a * lx * (1.f - ly);
            const float w01 = a * (1.f - lx) * ly;
            const float w11 = a * lx * ly;
            const bool vx0 = (x0 >= 0) & (x0 < Wl);
            const bool vx1 = (x1 >= 0) & (x1 < Wl);
            const bool vy0 = (y0 >= 0) & (y0 < Hl);
            const bool vy1 = (y1 >= 0) & (y1 < Hl);
            if (vx0 & vy0) acc += w00 * vlev[(size_t)(y0 * Wl + x0) * D_];
            if (vx1 & vy0) acc += w10 * vlev[(size_t)(y0 * Wl + x1) * D_];
            if (vx0 & vy1) acc += w01 * vlev[(size_t)(y1 * Wl + x0) * D_];
            if (vx1 & vy1) acc += w11 * vlev[(size_t)(y1 * Wl + x1) * D_];
        }
    }
    out[(size_t)tok * D_ + head * HD_ + lane] = acc;
}

// ---------------------------------------------------------------------------
// Y = LayerNorm(A + B) * g + be ; one wave per token (256 elems, 8/lane),
// wave32 __shfl_xor reductions.
// ---------------------------------------------------------------------------
__global__ __launch_bounds__(256) void add_ln_kernel(
    const float* __restrict__ A, const float* __restrict__ Bm,
    const float* __restrict__ g, const float* __restrict__ be,
    float* __restrict__ Y, int Ntok)
{
    const int lane = threadIdx.x & 31;
    const int t    = blockIdx.x * (blockDim.x >> 5) + (threadIdx.x >> 5);
    if (t >= Ntok) return;
    const float* __restrict__ ra = A  + (size_t)t * D_;
    const float* __restrict__ rb = Bm + (size_t)t * D_;
    float v[8];
    float s = 0.f;
#pragma unroll
    for (int i = 0; i < 8; ++i) { const int j = lane + 32 * i; v[i] = ra[j] + rb[j]; s += v[i]; }
#pragma unroll
    for (int o = 16; o > 0; o >>= 1) s += __shfl_xor(s, o, 32);
    const float mean = s * (1.f / 256.f);
    float vs = 0.f;
#pragma unroll
    for (int i = 0; i < 8; ++i) { const float d = v[i] - mean; vs += d * d; }
#pragma unroll
    for (int o = 16; o > 0; o >>= 1) vs += __shfl_xor(vs, o, 32);
    const float inv = rsqrtf(vs * (1.f / 256.f) + 1e-5f);
    float* __restrict__ ry = Y + (size_t)t * D_;
#pragma unroll
    for (int i = 0; i < 8; ++i) {
        const int j = lane + 32 * i;
        ry[j] = (v[i] - mean) * inv * g[j] + be[j];
    }
}

// ---------------------------------------------------------------------------
extern "C" void kernel_launch(void* const* d_in, const int* in_sizes, int n_in,
                              void* d_out, int out_size, void* d_ws, size_t ws_size,
                              hipStream_t stream)
{
    const float* src    = (const float*)d_in[0];
    const float* pos    = (const float*)d_in[1];
    const float* refp   = (const float*)d_in[2];
    const float* W_val  = (const float*)d_in[3];
    const float* b_val  = (const float*)d_in[4];
    const float* W_offs = (const float*)d_in[5];
    const float* b_offs = (const float*)d_in[6];
    const float* W_attn = (const float*)d_in[7];
    const float* b_attn = (const float*)d_in[8];
    const float* W_out  = (const float*)d_in[9];
    const float* b_out  = (const float*)d_in[10];
    const float* g1     = (const float*)d_in[11];
    const float* be1    = (const float*)d_in[12];
    const float* W_ff1  = (const float*)d_in[13];
    const float* b_ff1  = (const float*)d_in[14];
    const float* W_ff2  = (const float*)d_in[15];
    const float* b_ff2  = (const float*)d_in[16];
    const float* g2     = (const float*)d_in[17];
    const float* be2    = (const float*)d_in[18];
    float* out = (float*)d_out;

    const int Ntok = in_sizes[0] / D_;   // B * Lin = 43520
    const int Lq   = LIN_;               // fixed by SHAPES

    // Workspace layout (floats), with dead-buffer reuse:
    float* ws    = (float*)d_ws;
    float* value = ws;                                   // N*256
    float* offsb = value + (size_t)Ntok * D_;            // N*256  (reused as src2)
    float* attnb = offsb + (size_t)Ntok * D_;            // N*128
    float* samp  = attnb + (size_t)Ntok * 128;           // N*256  (reused as ff2 out)
    float* ff1   = samp  + (size_t)Ntok * D_;            // N*1024
    float* src2  = offsb;
    float* h     = value;
    float* ff2o  = samp;
    // Packed weights (~3 MB total), after activations:
    float* Pval  = ff1   + (size_t)Ntok * DFF_;          // 256*256
    float* Poffs = Pval  + 256 * 256;                    // 256*256
    float* Pattn = Poffs + 256 * 256;                    // 256*128
    float* Pout  = Pattn + 256 * 128;                    // 256*256
    float* Pff1  = Pout  + 256 * 256;                    // 256*1024
    float* Pff2  = Pff1  + 256 * 1024;                   // 1024*256

    // --- pack weights into K-pair-major (B fragment = one b64) ---
    pack_w_kernel<<<(256 * 256 + 255) / 256, 256, 0, stream>>>(W_val,  Pval,  256 * 256,  256);
    pack_w_kernel<<<(256 * 256 + 255) / 256, 256, 0, stream>>>(W_offs, Poffs, 256 * 256,  256);
    pack_w_kernel<<<(256 * 128 + 255) / 256, 256, 0, stream>>>(W_attn, Pattn, 256 * 128,  128);
    pack_w_kernel<<<(256 * 256 + 255) / 256, 256, 0, stream>>>(W_out,  Pout,  256 * 256,  256);
    pack_w_kernel<<<(256 * 1024 + 255) / 256, 256, 0, stream>>>(W_ff1, Pff1,  256 * 1024, 1024);
    pack_w_kernel<<<(1024 * 256 + 255) / 256, 256, 0, stream>>>(W_ff2, Pff2,  1024 * 256, 256);

    const int GB = 32 * WAVES_;      // 128 threads per GEMM block
    const int gx = Ntok / 32;        // RT=2 -> 32 rows per block (43520/32 = 1360)

    // Grid.y = NOUT / (WAVES_ * CT * 16)
    // 1) value = src @ W_val + b_val                    RT=2, CT=4
    gemm_wmma<false, false, 2, 4, 256, 256><<<dim3(gx, 1), GB, 0, stream>>>(src, nullptr, Pval, b_val, value);
    // 2) offs = (src+pos) @ W_offs + b_offs
    gemm_wmma<true, false, 2, 4, 256, 256><<<dim3(gx, 1), GB, 0, stream>>>(src, pos, Poffs, b_offs, offsb);
    // 3) attn logits = (src+pos) @ W_attn + b_attn      RT=2, CT=2 (128 cols)
    gemm_wmma<true, false, 2, 2, 256, 128><<<dim3(gx, 1), GB, 0, stream>>>(src, pos, Pattn, b_attn, attnb);
    // 4) softmax over NL*NP = 16
    {
        const int nrows = Ntok * NH_;
        softmax16_kernel<<<(nrows + 255) / 256, 256, 0, stream>>>(attnb, nrows);
    }
    // 5) deformable sampling -> samp [N, 256]
    {
        const int npairs = Ntok * NH_;
        msdeform_kernel<<<(npairs + 7) / 8, 256, 0, stream>>>(value, attnb, offsb, refp, samp, Ntok, Lq);
    }
    // 6) src2 = samp @ W_out + b_out   (src2 reuses offs buffer)
    gemm_wmma<false, false, 2, 4, 256, 256><<<dim3(gx, 1), GB, 0, stream>>>(samp, nullptr, Pout, b_out, src2);
    // 7) h = LN(src + src2)            (h reuses value buffer)
    add_ln_kernel<<<(Ntok + 7) / 8, 256, 0, stream>>>(src, src2, g1, be1, h, Ntok);
    // 8) ff1 = relu(h @ W_ff1 + b_ff1)                  grid.y = 1024/256 = 4
    gemm_wmma<false, true, 2, 4, 256, 1024><<<dim3(gx, 4), GB, 0, stream>>>(h, nullptr, Pff1, b_ff1, ff1);
    // 9) ff2 = ff1 @ W_ff2 + b_ff2     (reuses samp buffer)
    gemm_wmma<false, false, 2, 4, 1024, 256><<<dim3(gx, 1), GB, 0, stream>>>(ff1, nullptr, Pff2, b_ff2, ff2o);
    // 10) out = LN(h + ff2)
    add_ln_kernel<<<(Ntok + 7) / 8, 256, 0, stream>>>(h, ff2o, g2, be2, out, Ntok);
}